// DynamicsModel_85469849190529
// MI455X (gfx1250) — compile-verified
//
#include <hip/hip_runtime.h>
#include <stdint.h>

// ---------------------------------------------------------------------------
// out = -0.1*(deg*x - Ax) + 0.9*hyper_res
// Per adj edge (s,d):      out[d] += 0.1f*(x[s] - x[d])
// Per hyper edge (a,b,c):  p = x[b]*x[c];  out[k] += 0.9f*(p - x[k]^2), k in {a,b,c}
// Roofline: HBM traffic = 188 MB of edge streams (~8us @ 23.3 TB/s); x (2MB)
// and out (2MB) live in the 192MB L2, so gathers/atomics are L2-resident.
// Streams are loaded non-temporally to keep x/out hot in L2.
// ---------------------------------------------------------------------------

typedef uint32_t u32x4 __attribute__((ext_vector_type(4)));

__global__ __launch_bounds__(256) void zero_kernel(float* __restrict__ out, int n) {
    int i = blockIdx.x * blockDim.x + threadIdx.x;
    if (i < n) out[i] = 0.0f;
}

// Adjacency scatter: stream (src,dst) pairs two-at-a-time via 16B non-temporal
// loads, gather x from L2, one non-returning f32 atomic per edge.
__global__ __launch_bounds__(256) void adj_kernel(const float* __restrict__ x,
                                                  const uint32_t* __restrict__ edges, // 2 ints/edge
                                                  float* __restrict__ out, int n_edges) {
    int n_pairs = n_edges >> 1;
    int stride  = gridDim.x * blockDim.x;
    for (int i = blockIdx.x * blockDim.x + threadIdx.x; i < n_pairs; i += stride) {
        u32x4 e = __builtin_nontemporal_load((const u32x4*)edges + i);
        int s0 = (int)e.x, d0 = (int)e.y;
        int s1 = (int)e.z, d1 = (int)e.w;
        float v0 = 0.1f * (x[s0] - x[d0]);
        float v1 = 0.1f * (x[s1] - x[d1]);
        atomicAdd(out + d0, v0);   // no-return global_atomic_add_f32
        atomicAdd(out + d1, v1);
    }
    // tail (n_edges odd) — handled by thread 0 of block 0
    if ((n_edges & 1) && blockIdx.x == 0 && threadIdx.x == 0) {
        int s = (int)edges[2 * (n_edges - 1) + 0];
        int d = (int)edges[2 * (n_edges - 1) + 1];
        atomicAdd(out + d, 0.1f * (x[s] - x[d]));
    }
}

// ---------------------------------------------------------------------------
// Hyper scatter with CDNA5 async Global->LDS staging.
// Tile = 1024 edges = 12288 bytes. 256 threads * 3 chunks * 16B = 12288B.
// Double-buffered: issue tile t+1, s_wait_asynccnt for tile t, barrier, consume.
// ---------------------------------------------------------------------------
#define TILE_EDGES   1024
#define TILE_BYTES   (TILE_EDGES * 12)     // 12288
#define TILE_INTS    (TILE_BYTES / 4)      // 3072
#define CHUNKS       3                     // 16B async chunks per thread

static __device__ __forceinline__ uint32_t lds_offset_u32(const void* p) {
    // Generic LDS pointer = {shared_aperture_hi32, lds_byte_offset}; low 32 bits
    // are the wave-relative LDS byte address used by DS / async-to-LDS ops.
    return (uint32_t)(uintptr_t)p;
}

static __device__ __forceinline__ void issue_tile_async(uint32_t lds_base,
                                                        uint64_t gbase,
                                                        uint64_t tile_byte,
                                                        uint64_t total_bytes,
                                                        int tid) {
#pragma unroll
    for (int k = 0; k < CHUNKS; ++k) {
        uint32_t local = (uint32_t)(k * 4096 + tid * 16);
        uint64_t goff  = tile_byte + (uint64_t)local;
        // Clamp OOB chunks into bounds (consumed garbage is guarded later).
        if (goff + 16 > total_bytes) goff = total_bytes - 16;
        uint64_t gaddr = gbase + goff;
        uint32_t laddr = lds_base + local;
        asm volatile("global_load_async_to_lds_b128 %0, %1, off th:TH_LOAD_NT"
                     :: "v"(laddr), "v"(gaddr)
                     : "memory");
    }
}

__global__ __launch_bounds__(256) void hyper_kernel(const float* __restrict__ x,
                                                    const int* __restrict__ he, // 3*n_hyper ints
                                                    float* __restrict__ out,
                                                    int n_hyper) {
    __shared__ __align__(16) int tile_lds[2][TILE_INTS];   // 24 KB of the 320 KB WGP LDS

    const int tid        = threadIdx.x;
    const int num_tiles  = (n_hyper + TILE_EDGES - 1) / TILE_EDGES;
    const uint64_t gbase = (uint64_t)(uintptr_t)he;
    const uint64_t total_bytes = (uint64_t)n_hyper * 12ull;

    int tile = blockIdx.x;
    if (tile >= num_tiles) return;   // uniform per block

    const uint32_t lbase[2] = { lds_offset_u32(&tile_lds[0][0]),
                                lds_offset_u32(&tile_lds[1][0]) };

    int cur = 0;
    issue_tile_async(lbase[0], gbase, (uint64_t)tile * TILE_BYTES, total_bytes, tid);

    while (tile < num_tiles) {
        int  next      = tile + gridDim.x;
        bool have_next = (next < num_tiles);
        if (have_next) {
            issue_tile_async(lbase[cur ^ 1], gbase,
                             (uint64_t)next * TILE_BYTES, total_bytes, tid);
            asm volatile("s_wait_asynccnt 3" ::: "memory"); // current tile's 3 done
        } else {
            asm volatile("s_wait_asynccnt 0" ::: "memory");
        }
        __syncthreads();   // all waves' staging for `cur` visible

        const int* tl = &tile_lds[cur][0];
        int ebase = tile * TILE_EDGES;
#pragma unroll
        for (int j = 0; j < TILE_EDGES / 256; ++j) {
            int el = tid * (TILE_EDGES / 256) + j;
            int eg = ebase + el;
            if (eg < n_hyper) {
                int i0 = tl[el * 3 + 0];
                int i1 = tl[el * 3 + 1];
                int i2 = tl[el * 3 + 2];
                float x0 = x[i0], x1 = x[i1], x2 = x[i2];
                float p  = x1 * x2;
                atomicAdd(out + i0, 0.9f * (p - x0 * x0));
                atomicAdd(out + i1, 0.9f * (p - x1 * x1));
                atomicAdd(out + i2, 0.9f * (p - x2 * x2));
            }
        }
        __syncthreads();   // done reading `cur` before it gets overwritten next iter
        cur ^= 1;
        tile = next;
    }
}

// ---------------------------------------------------------------------------
extern "C" void kernel_launch(void* const* d_in, const int* in_sizes, int n_in,
                              void* d_out, int out_size, void* d_ws, size_t ws_size,
                              hipStream_t stream) {
    (void)n_in; (void)d_ws; (void)ws_size;
    const float*    x  = (const float*)d_in[0];
    // d_in[1] = t (unused scalar)
    const int*      he = (const int*)d_in[2];      // 3 * n_hyper
    const uint32_t* ae = (const uint32_t*)d_in[3]; // 2 ints per edge

    float* out   = (float*)d_out;
    int n_nodes  = in_sizes[0];
    int n_hyper  = in_sizes[2] / 3;
    int n_adj    = in_sizes[3] / 2;

    zero_kernel<<<(n_nodes + 255) / 256, 256, 0, stream>>>(out, n_nodes);

    // 2 edges per 16B load, ~4 pairs per thread
    int adj_blocks = ((n_adj / 2) + 256 * 4 - 1) / (256 * 4);
    if (adj_blocks < 1) adj_blocks = 1;
    adj_kernel<<<adj_blocks, 256, 0, stream>>>(x, ae, out, n_adj);

    int num_tiles = (n_hyper + TILE_EDGES - 1) / TILE_EDGES;
    hyper_kernel<<<num_tiles, 256, 0, stream>>>(x, he, out, n_hyper);
}